// Messages_27127013442039
// MI455X (gfx1250) — compile-verified
//
#include <hip/hip_runtime.h>
#include <cstddef>

typedef __attribute__((ext_vector_type(2))) float v2f;
typedef __attribute__((ext_vector_type(8))) float v8f;

// ---- fp32 WMMA 16x16x4: D = A(16x4) * B(4x16) + C ----------------------
// A frag: lane<16 -> row m=lane, k={k0,k0+1}; lane>=16 -> k={k0+2,k0+3}
// B frag: lane<16 -> col n=lane, k={k0,k0+1}; lane>=16 -> k={k0+2,k0+3}
// D: vgpr v on lane L -> row v+8*(L>>4), col L&15
__device__ __forceinline__ v8f wmma4(v2f a, v2f b, v8f c) {
  return __builtin_amdgcn_wmma_f32_16x16x4_f32(false, a, false, b, (short)0, c,
                                               false, false);
}

// One 16x16 output tile of   out[m][n0+n] += sum_k A[m][k] * W[n0+n][k]
// A: row-major [16 x >=K] with leading dim ldA (LDS or global)
// W: row-major [Nout x ldW] weight matrix (acts as B^T), global
template <int K>
__device__ __forceinline__ v8f gemm_seg(const float* A, int ldA,
                                        const float* __restrict__ W, int ldW,
                                        int n0, v8f acc, int l15, int h) {
  const float* arow = A + l15 * ldA + 2 * h;
  const float* wrow = W + (size_t)(n0 + l15) * ldW + 2 * h;
#pragma unroll
  for (int k = 0; k < K; k += 4) {
    v2f a = *(const v2f*)(arow + k);
    v2f b = *(const v2f*)(wrow + k);
    acc = wmma4(a, b, acc);
  }
  return acc;
}

// ======================= Phase 1: node features ==========================
// l_a_node[n][c] = sum_d x_a[n][d] * W_L0[c][d]            (N x 32, K=64)
// l_v_node[n][c][i] = sum_d x_v[n][d][i] * W_L1[c][d]      (N x 32 x 3, K=16)
__global__ __launch_bounds__(128) void node_feats_kernel(
    const float* __restrict__ xa, const float* __restrict__ xv,
    const float* __restrict__ W_L0, const float* __restrict__ W_L1,
    float* __restrict__ la_node, float* __restrict__ lv_node, int N) {
  const int wave = threadIdx.x >> 5;
  const int lane = threadIdx.x & 31;
  const int l15 = lane & 15;
  const int h = lane >> 4;
  const int r0 = (blockIdx.x * 4 + wave) * 16;
  if (r0 >= N) return;  // wave-uniform, no barriers in this kernel
  const int mrow = r0 + l15;
  const int mc = mrow < N ? mrow : N - 1;

  // l_a  : 2 tiles of 16x16, K=64
  const float* arow = xa + (size_t)mc * 64 + 2 * h;
#pragma unroll
  for (int t = 0; t < 2; ++t) {
    v8f acc = {};
    const float* wrow = W_L0 + (size_t)(16 * t + l15) * 64 + 2 * h;
#pragma unroll
    for (int k = 0; k < 64; k += 4) {
      v2f a = *(const v2f*)(arow + k);
      v2f b = *(const v2f*)(wrow + k);
      acc = wmma4(a, b, acc);
    }
#pragma unroll
    for (int v = 0; v < 8; ++v) {
      int m = r0 + v + 8 * h;
      if (m < N) la_node[(size_t)m * 32 + 16 * t + l15] = acc[v];
    }
  }

  // l_v : per component i, 2 tiles of 16x16, K=16
  const float* abase = xv + (size_t)mc * 48;
#pragma unroll
  for (int i = 0; i < 3; ++i) {
#pragma unroll
    for (int t = 0; t < 2; ++t) {
      v8f acc = {};
      const float* wrow = W_L1 + (size_t)(16 * t + l15) * 16 + 2 * h;
#pragma unroll
      for (int k = 0; k < 16; k += 4) {
        v2f a;
        a.x = abase[(k + 2 * h) * 3 + i];
        a.y = abase[(k + 2 * h + 1) * 3 + i];
        v2f b = *(const v2f*)(wrow + k);
        acc = wmma4(a, b, acc);
      }
#pragma unroll
      for (int v = 0; v < 8; ++v) {
        int m = r0 + v + 8 * h;
        if (m < N) lv_node[(size_t)m * 96 + (size_t)(16 * t + l15) * 3 + i] = acc[v];
      }
    }
  }
}

// ======================= Phase 2: edge kernel ============================
#define LDA 98  // 96+2 pad (bank scramble), also used for 64-wide panels
#define LDO 66  // 64+2 pad

struct __align__(16) WaveBuf {
  float A[16 * LDA];   // GEMM input panel (up to 16 x 96)
  float O[16 * LDO];   // GEMM output staging (16 x 64)
  float phia[16 * 32]; // phi_a
  float la[16 * 32];   // gathered l_a
  float lv[16 * 96];   // gathered l_v, [e][c*3+i]
  float rh[16 * 4];    // squashed edge vector + dist
  int srcI[16];
  int dstI[16];
};

__global__ __launch_bounds__(64) void edge_kernel(
    const float* __restrict__ r_ij, const int* __restrict__ src,
    const int* __restrict__ dst, const float* __restrict__ la_node,
    const float* __restrict__ lv_node, const float* __restrict__ W_enc,
    const float* __restrict__ b_enc, const float* __restrict__ W_Y000,
    const float* __restrict__ W_Y110, const float* __restrict__ W_Y011,
    const float* __restrict__ W_Y101, const float* __restrict__ W_Y111,
    const float* __restrict__ W_m1, const float* __restrict__ b_m1,
    const float* __restrict__ W_m2, const float* __restrict__ b_m2,
    const float* __restrict__ W_m3, float* __restrict__ Ba,
    float* __restrict__ Bv, int E) {
  __shared__ WaveBuf S[2];  // 2 waves / block
  const int wave = threadIdx.x >> 5;
  const int lane = threadIdx.x & 31;
  const int l15 = lane & 15;
  const int h = lane >> 4;
  WaveBuf& B = S[wave];
  const int e0 = (blockIdx.x * 2 + wave) * 16;

  // ---- step 0: edge geometry, indices (lanes 0..15) ----
  if (lane < 16) {
    size_t e = (size_t)e0 + lane;
    size_t ec = (e < (size_t)E) ? e : (size_t)(E - 1);
    float x = r_ij[ec * 3 + 0], y = r_ij[ec * 3 + 1], z = r_ij[ec * 3 + 2];
    float d = sqrtf(x * x + y * y + z * z);
    float s = 1.4f * d;                       // (7/R0)*|r|
    float f = tanhf(s) / fmaxf(s, 1e-12f);    // tens_sigmoid1 factor
    B.rh[lane * 4 + 0] = 1.4f * x * f;
    B.rh[lane * 4 + 1] = 1.4f * y * f;
    B.rh[lane * 4 + 2] = 1.4f * z * f;
    B.rh[lane * 4 + 3] = d;
    B.srcI[lane] = src[ec];
    B.dstI[lane] = dst[ec];
  }
  __syncthreads();

  // ---- step 1: radial encoding -> A[:, 0:64]  (cos | sin) ----
  {
    float d = B.rh[l15 * 4 + 3];
    const float c0 = 0.62831853071795864769f;  // pi / 5
#pragma unroll
    for (int j = 0; j < 32; ++j) {
      float p = c0 * (float)(1 + (j >> 1)) * d;
      B.A[l15 * LDA + 32 * h + j] = h ? sinf(p) : cosf(p);
    }
  }
  __syncthreads();

  // ---- step 2: phi_a = rad @ W_enc^T + b_enc   (16x32, K=64) ----
#pragma unroll
  for (int t = 0; t < 2; ++t) {
    v8f acc = {};
    acc = gemm_seg<64>(B.A, LDA, W_enc, 64, 16 * t, acc, l15, h);
    float bias = b_enc[16 * t + l15];
#pragma unroll
    for (int v = 0; v < 8; ++v)
      B.phia[(v + 8 * h) * 32 + 16 * t + l15] = acc[v] + bias;
  }
  __syncthreads();

  // ---- step 3: gather l_a (16x32) and l_v (16x96) at dst ----
  {
    size_t row = (size_t)B.dstI[l15];
    const float4* pa = (const float4*)(la_node + row * 32 + 16 * h);
    float4* qa = (float4*)(B.la + l15 * 32 + 16 * h);
#pragma unroll
    for (int q = 0; q < 4; ++q) qa[q] = pa[q];
    const float4* pv = (const float4*)(lv_node + row * 96 + 48 * h);
    float4* qv = (float4*)(B.lv + l15 * 96 + 48 * h);
#pragma unroll
    for (int q = 0; q < 12; ++q) qv[q] = pv[q];
  }
  __syncthreads();

  // ---- step 4: A = [ y000 | y110 ]  (16 x 64) ----
#pragma unroll
  for (int cc = 0; cc < 16; ++cc) {
    int c = 16 * h + cc;
    float pa = B.phia[l15 * 32 + c];
    float lav = B.la[l15 * 32 + c];
    float v0 = B.lv[l15 * 96 + c * 3 + 0];
    float v1 = B.lv[l15 * 96 + c * 3 + 1];
    float v2 = B.lv[l15 * 96 + c * 3 + 2];
    float dotv = v0 * B.rh[l15 * 4 + 0] + v1 * B.rh[l15 * 4 + 1] +
                 v2 * B.rh[l15 * 4 + 2];
    B.A[l15 * LDA + c] = lav * pa;             // y000
    B.A[l15 * LDA + 32 + c] = pa * dotv;       // y110 (phi_v = pa*rh)
  }
  __syncthreads();

  // ---- step 5: psi_a = y000 @ W_Y000^T + y110 @ W_Y110^T  (16x64) ----
  v8f psi[4];
#pragma unroll
  for (int t = 0; t < 4; ++t) {
    v8f acc = {};
    acc = gemm_seg<32>(B.A, LDA, W_Y000, 32, 16 * t, acc, l15, h);
    acc = gemm_seg<32>(B.A + 32, LDA, W_Y110, 32, 16 * t, acc, l15, h);
    psi[t] = acc;
#pragma unroll
    for (int v = 0; v < 8; ++v)
      B.O[(v + 8 * h) * LDO + 16 * t + l15] = acc[v];
  }
  __syncthreads();

  // ---- step 6: residual MLP (3x 64x64 GEMM, leaky 0.1) ----
#pragma unroll
  for (int t = 0; t < 4; ++t) {
    v8f acc = {};
    acc = gemm_seg<64>(B.O, LDO, W_m1, 64, 16 * t, acc, l15, h);
    float bias = b_m1[16 * t + l15];
#pragma unroll
    for (int v = 0; v < 8; ++v) {
      float x = acc[v] + bias;
      B.A[(v + 8 * h) * LDA + 16 * t + l15] = x > 0.f ? x : 0.1f * x;
    }
  }
  __syncthreads();
#pragma unroll
  for (int t = 0; t < 4; ++t) {
    v8f acc = {};
    acc = gemm_seg<64>(B.A, LDA, W_m2, 64, 16 * t, acc, l15, h);
    float bias = b_m2[16 * t + l15];
#pragma unroll
    for (int v = 0; v < 8; ++v) {
      float x = acc[v] + bias;
      B.O[(v + 8 * h) * LDO + 16 * t + l15] = x > 0.f ? x : 0.1f * x;
    }
  }
  __syncthreads();
#pragma unroll
  for (int t = 0; t < 4; ++t) {
    v8f acc = {};
    acc = gemm_seg<64>(B.O, LDO, W_m3, 64, 16 * t, acc, l15, h);
#pragma unroll
    for (int v = 0; v < 8; ++v) {
      int m = v + 8 * h;
      int e = e0 + m;
      if (e < E) {
        float val = 0.1f * (psi[t][v] + acc[v]);  // residual + output scale
        atomicAdd(&Ba[(size_t)B.srcI[m] * 64 + 16 * t + l15], val);
      }
    }
  }
  __syncthreads();

  // ---- step 7: psi_v per component: [y011|y101|y111] @ stacked W (K=96) ----
  const int i1v[3] = {1, 2, 0};
  const int i2v[3] = {2, 0, 1};
  for (int i = 0; i < 3; ++i) {
    int i1 = i1v[i], i2 = i2v[i];
    {
      float r_i = B.rh[l15 * 4 + i];
      float r_a = B.rh[l15 * 4 + i1];
      float r_b = B.rh[l15 * 4 + i2];
#pragma unroll
      for (int cc = 0; cc < 16; ++cc) {
        int c = 16 * h + cc;
        float pa = B.phia[l15 * 32 + c];
        float lav = B.la[l15 * 32 + c];
        float vi = B.lv[l15 * 96 + c * 3 + i];
        float va = B.lv[l15 * 96 + c * 3 + i1];
        float vb = B.lv[l15 * 96 + c * 3 + i2];
        B.A[l15 * LDA + c] = lav * pa * r_i;                  // y011
        B.A[l15 * LDA + 32 + c] = pa * vi;                    // y101
        B.A[l15 * LDA + 64 + c] = pa * (va * r_b - vb * r_a); // y111 = cross
      }
    }
    __syncthreads();
    v8f acc = {};
    acc = gemm_seg<32>(B.A, LDA, W_Y011, 32, 0, acc, l15, h);
    acc = gemm_seg<32>(B.A + 32, LDA, W_Y101, 32, 0, acc, l15, h);
    acc = gemm_seg<32>(B.A + 64, LDA, W_Y111, 32, 0, acc, l15, h);
#pragma unroll
    for (int v = 0; v < 8; ++v) {
      int m = v + 8 * h;
      int e = e0 + m;
      if (e < E)
        atomicAdd(&Bv[(size_t)B.srcI[m] * 48 + (size_t)l15 * 3 + i],
                  0.1f * acc[v]);
    }
    __syncthreads();
  }
}

// ============================ launch =====================================
extern "C" void kernel_launch(void* const* d_in, const int* in_sizes, int n_in,
                              void* d_out, int out_size, void* d_ws,
                              size_t ws_size, hipStream_t stream) {
  const float* r_ij = (const float*)d_in[0];
  const float* x_a = (const float*)d_in[1];
  const float* x_v = (const float*)d_in[2];
  const int* src = (const int*)d_in[3];
  const int* dst = (const int*)d_in[4];
  const float* W_L0 = (const float*)d_in[5];
  const float* W_L1 = (const float*)d_in[6];
  const float* W_enc = (const float*)d_in[7];
  const float* b_enc = (const float*)d_in[8];
  const float* W_Y000 = (const float*)d_in[9];
  const float* W_Y110 = (const float*)d_in[10];
  const float* W_Y011 = (const float*)d_in[11];
  const float* W_Y101 = (const float*)d_in[12];
  const float* W_Y111 = (const float*)d_in[13];
  const float* W_m1 = (const float*)d_in[14];
  const float* b_m1 = (const float*)d_in[15];
  const float* W_m2 = (const float*)d_in[16];
  const float* b_m2 = (const float*)d_in[17];
  const float* W_m3 = (const float*)d_in[18];

  const int E = in_sizes[3];       // src count
  const int N = in_sizes[1] / 64;  // x_a rows

  hipMemsetAsync(d_out, 0, (size_t)out_size * sizeof(float), stream);

  size_t need = (size_t)N * (32 + 96) * sizeof(float);
  if (ws_size < need) return;  // outputs stay zero (deterministic fallback)

  float* la_node = (float*)d_ws;                 // N x 32
  float* lv_node = la_node + (size_t)N * 32;     // N x 32 x 3
  float* Ba = (float*)d_out;                     // N x 64
  float* Bv = Ba + (size_t)N * 64;               // N x 16 x 3

  int nb_node = (N + 63) / 64;  // 128 thr = 4 waves x 16 rows
  node_feats_kernel<<<nb_node, 128, 0, stream>>>(x_a, x_v, W_L0, W_L1, la_node,
                                                 lv_node, N);

  int nb_edge = (E + 31) / 32;  // 64 thr = 2 waves x 16 edges
  edge_kernel<<<nb_edge, 64, 0, stream>>>(
      r_ij, src, dst, la_node, lv_node, W_enc, b_enc, W_Y000, W_Y110, W_Y011,
      W_Y101, W_Y111, W_m1, b_m1, W_m2, b_m2, W_m3, Ba, Bv, E);
}